// AevaNaiveMoe_33414845563491
// MI455X (gfx1250) — compile-verified
//
#include <hip/hip_runtime.h>
#include <math.h>

// ---------------- CDNA5 WMMA (fp32, 16x16x4) ----------------
typedef __attribute__((ext_vector_type(2))) float v2f;
typedef __attribute__((ext_vector_type(8))) float v8f;

#define HID 1024
#define NEXP 32
#define NLG 16
#define TOPK 6
#define TTOK 1024
#define CAP 1024

__device__ __forceinline__ v8f wmma4(v2f a, v2f b, v8f c) {
  // D = A(16x4 f32) * B(4x16 f32) + C(16x16 f32)
  return __builtin_amdgcn_wmma_f32_16x16x4_f32(false, a, false, b, (short)0, c,
                                               false, false);
}

// ---------------- Router: scores + group-limited top-k + scatter ----------------
__global__ __launch_bounds__(32) void router_kernel(
    const float* __restrict__ x, const float* __restrict__ gw,
    const float* __restrict__ gb, int* __restrict__ ecnt,
    int* __restrict__ etok, int* __restrict__ epair, float* __restrict__ ew,
    int* __restrict__ acnt, int* __restrict__ atok) {
  int t = blockIdx.x;
  int e = threadIdx.x;  // 32 lanes == 32 experts
  const float4* xr = (const float4*)(x + (size_t)t * HID);
  const float4* wr = (const float4*)(gw + (size_t)e * HID);
  float acc = 0.f;
#pragma unroll 4
  for (int i = 0; i < HID / 4; ++i) {
    float4 a = xr[i], b = wr[i];
    acc += a.x * b.x + a.y * b.y + a.z * b.z + a.w * b.w;
  }
  float score = 1.f / (1.f + __expf(-acc));  // sigmoid
  float sc = score + gb[e];                  // score_for_choice
  __shared__ float s_sc[NEXP];
  __shared__ float s_score[NEXP];
  s_sc[e] = sc;
  s_score[e] = score;
  __syncthreads();
  if (e == 0) {
    // group scores: 4 groups of 8, sum of top-2
    float gsc[4];
#pragma unroll
    for (int g = 0; g < 4; ++g) {
      float m1 = -1e30f, m2 = -1e30f;
#pragma unroll
      for (int j = 0; j < 8; ++j) {
        float v = s_sc[g * 8 + j];
        if (v > m1) { m2 = m1; m1 = v; }
        else if (v > m2) { m2 = v; }
      }
      gsc[g] = m1 + m2;
    }
    // top-2 groups (earliest index wins ties, like lax.top_k)
    int g0 = 0;
    for (int g = 1; g < 4; ++g) if (gsc[g] > gsc[g0]) g0 = g;
    int g1 = -1;
    for (int g = 0; g < 4; ++g) {
      if (g == g0) continue;
      if (g1 < 0 || gsc[g] > gsc[g1]) g1 = g;
    }
    unsigned allowed = (0xFFu << (g0 * 8)) | (0xFFu << (g1 * 8));
    // top-6 among allowed experts
    int id[TOPK];
    unsigned chosen = 0;
    for (int k = 0; k < TOPK; ++k) {
      int best = -1;
      float bv = -1e30f;
      for (int e2 = 0; e2 < NEXP; ++e2) {
        if (!((allowed >> e2) & 1u) || ((chosen >> e2) & 1u)) continue;
        float v = s_sc[e2];
        if (v > bv) { bv = v; best = e2; }
      }
      id[k] = best;
      chosen |= 1u << best;
    }
    float wv[TOPK];
    float wsum = 0.f;
    for (int k = 0; k < TOPK; ++k) { wv[k] = s_score[id[k]]; wsum += wv[k]; }
    float scale = 0.2f / (wsum + 1e-20f);  // ROUTED_SCALE
    for (int k = 0; k < TOPK; ++k) {
      int ex = id[k];
      int s = atomicAdd(&ecnt[ex], 1);
      etok[ex * CAP + s] = t;
      epair[ex * CAP + s] = t * TOPK + k;
      ew[ex * CAP + s] = wv[k] * scale;
    }
    int lg = id[0] >> 1;  // E/L = 2
    int s2 = atomicAdd(&acnt[lg], 1);
    atok[lg * CAP + s2] = t;
  }
}

// ---------------- GEMM1: h = silu(X*Wg) * (X*Wu), 16x64 tile per wave ----------------
__global__ __launch_bounds__(32) void gemm1_kernel(
    const float* __restrict__ x, const float* __restrict__ wgAll,
    const float* __restrict__ wuAll, const int* __restrict__ cnt,
    const int* __restrict__ tokList, const int* __restrict__ pairList,
    float* __restrict__ hOut, int I, int fixedCount) {
  int e = blockIdx.z;
  int m0 = blockIdx.y << 4;
  int n0 = blockIdx.x << 6;
  int count = cnt ? cnt[e] : fixedCount;
  if (m0 >= count) return;
  int lane = threadIdx.x;
  int r = lane & 15, kp = lane >> 4;

  int slotA = m0 + r;
  if (slotA >= count) slotA = count - 1;
  int tokA = tokList ? tokList[e * CAP + slotA] : slotA;
  const float* arow = x + (size_t)tokA * HID + 2 * kp;

  const float* wg = wgAll + (size_t)e * HID * I + (size_t)(2 * kp) * I + n0 + r;
  const float* wu = wuAll + (size_t)e * HID * I + (size_t)(2 * kp) * I + n0 + r;

  v8f accg[4] = {};
  v8f accu[4] = {};
  for (int k = 0; k < HID; k += 4) {
    v2f a = *(const v2f*)(arow + k);
    size_t bo = (size_t)k * I;
#pragma unroll
    for (int j = 0; j < 4; ++j) {
      v2f bg;
      bg.x = wg[bo + j * 16];
      bg.y = wg[bo + I + j * 16];
      accg[j] = wmma4(a, bg, accg[j]);
      v2f bu2;
      bu2.x = wu[bo + j * 16];
      bu2.y = wu[bo + I + j * 16];
      accu[j] = wmma4(a, bu2, accu[j]);
    }
  }
  int rowIdx[8];
#pragma unroll
  for (int v = 0; v < 8; ++v) {
    int gs = m0 + 8 * kp + v;
    if (gs < count) {
      int tk = tokList ? tokList[e * CAP + gs] : gs;
      rowIdx[v] = pairList ? pairList[e * CAP + gs] : tk;
    } else {
      rowIdx[v] = -1;
    }
  }
#pragma unroll
  for (int j = 0; j < 4; ++j) {
#pragma unroll
    for (int v = 0; v < 8; ++v) {
      if (rowIdx[v] >= 0) {
        float g = accg[j][v], u = accu[j][v];
        float hv = (g / (1.f + __expf(-g))) * u;  // silu(g) * u
        hOut[(size_t)rowIdx[v] * I + n0 + j * 16 + r] = hv;
      }
    }
  }
}

// ---------------- GEMM2: out = (h @ Wdown) * rowWeight ----------------
__global__ __launch_bounds__(32) void gemm2_kernel(
    const float* __restrict__ hIn, const float* __restrict__ wdAll,
    const int* __restrict__ cnt, const int* __restrict__ tokList,
    const int* __restrict__ pairList, const float* __restrict__ wlist,
    float* __restrict__ outBuf, int I, int fixedCount, float constScale) {
  int e = blockIdx.z;
  int m0 = blockIdx.y << 4;
  int n0 = blockIdx.x << 6;
  int count = cnt ? cnt[e] : fixedCount;
  if (m0 >= count) return;
  int lane = threadIdx.x;
  int r = lane & 15, kp = lane >> 4;

  int slotA = m0 + r;
  if (slotA >= count) slotA = count - 1;
  int tokA = tokList ? tokList[e * CAP + slotA] : slotA;
  int rowA = pairList ? pairList[e * CAP + slotA] : tokA;
  const float* arow = hIn + (size_t)rowA * I + 2 * kp;

  const float* wd = wdAll + (size_t)e * I * HID + (size_t)(2 * kp) * HID + n0 + r;

  v8f acc[4] = {};
  for (int k = 0; k < I; k += 4) {
    v2f a = *(const v2f*)(arow + k);
    size_t bo = (size_t)k * HID;
#pragma unroll
    for (int j = 0; j < 4; ++j) {
      v2f b;
      b.x = wd[bo + j * 16];
      b.y = wd[bo + HID + j * 16];
      acc[j] = wmma4(a, b, acc[j]);
    }
  }
  int rowIdx[8];
  float rowW[8];
#pragma unroll
  for (int v = 0; v < 8; ++v) {
    int gs = m0 + 8 * kp + v;
    if (gs < count) {
      int tk = tokList ? tokList[e * CAP + gs] : gs;
      rowIdx[v] = pairList ? pairList[e * CAP + gs] : tk;
      rowW[v] = wlist ? wlist[e * CAP + gs] : constScale;
    } else {
      rowIdx[v] = -1;
      rowW[v] = 0.f;
    }
  }
#pragma unroll
  for (int j = 0; j < 4; ++j) {
#pragma unroll
    for (int v = 0; v < 8; ++v) {
      if (rowIdx[v] >= 0)
        outBuf[(size_t)rowIdx[v] * HID + n0 + j * 16 + r] = acc[j][v] * rowW[v];
    }
  }
}

// ---------------- combine: out = shared + 0.1*anchor + sum_k routed_k ----------------
__global__ void combine_kernel(const float* __restrict__ sh,
                               const float* __restrict__ an,
                               const float* __restrict__ ro,
                               float* __restrict__ out) {
  int idx = blockIdx.x * 256 + threadIdx.x;
  int t = idx >> 10;
  int h = idx & (HID - 1);
  float v = sh[idx] + an[idx];  // anchor already scaled by 0.1
#pragma unroll
  for (int k = 0; k < TOPK; ++k)
    v += ro[(size_t)(t * TOPK + k) * HID + h];  // already scaled by tw
  out[idx] = v;
}

__global__ void init_counts(int* __restrict__ ecnt, int* __restrict__ acnt) {
  int i = threadIdx.x;
  if (i < NEXP) ecnt[i] = 0;
  if (i < NLG) acnt[i] = 0;
}

// ---------------- host launcher ----------------
extern "C" void kernel_launch(void* const* d_in, const int* in_sizes, int n_in,
                              void* d_out, int out_size, void* d_ws,
                              size_t ws_size, hipStream_t stream) {
  (void)in_sizes; (void)n_in; (void)out_size; (void)ws_size;
  const float* x       = (const float*)d_in[0];
  const float* gate_w  = (const float*)d_in[1];
  const float* gate_b  = (const float*)d_in[2];
  const float* w_gate  = (const float*)d_in[3];
  const float* w_up    = (const float*)d_in[4];
  const float* w_down  = (const float*)d_in[5];
  const float* sw_gate = (const float*)d_in[6];
  const float* sw_up   = (const float*)d_in[7];
  const float* sw_down = (const float*)d_in[8];
  const float* aw_gate = (const float*)d_in[9];
  const float* aw_up   = (const float*)d_in[10];
  const float* aw_down = (const float*)d_in[11];

  char* ws = (char*)d_ws;
  size_t off = 0;
  auto carve = [&](size_t bytes) -> void* {
    size_t o = (off + 255) & ~(size_t)255;
    off = o + bytes;
    return (void*)(ws + o);
  };
  int*   ecnt       = (int*)carve(NEXP * 4);
  int*   acnt       = (int*)carve(NLG * 4);
  int*   etok       = (int*)carve((size_t)NEXP * CAP * 4);
  int*   epair      = (int*)carve((size_t)NEXP * CAP * 4);
  float* ew         = (float*)carve((size_t)NEXP * CAP * 4);
  int*   atok       = (int*)carve((size_t)NLG * CAP * 4);
  float* h_pair     = (float*)carve((size_t)TTOK * TOPK * 512 * 4);   // 12.6 MB
  float* routed_out = (float*)carve((size_t)TTOK * TOPK * HID * 4);   // 25.2 MB
  float* anchor_h   = (float*)carve((size_t)TTOK * 256 * 4);          // 1 MB
  float* anchor_out = (float*)carve((size_t)TTOK * HID * 4);          // 4 MB
  float* shared_h   = (float*)carve((size_t)TTOK * 512 * 4);          // 2 MB
  float* shared_out = (float*)carve((size_t)TTOK * HID * 4);          // 4 MB

  init_counts<<<1, 32, 0, stream>>>(ecnt, acnt);
  router_kernel<<<TTOK, 32, 0, stream>>>(x, gate_w, gate_b, ecnt, etok, epair,
                                         ew, acnt, atok);
  // routed experts: I=512 -> 8 n-tiles of 64; up to 64 m-tiles; 32 experts
  gemm1_kernel<<<dim3(8, 64, NEXP), 32, 0, stream>>>(
      x, w_gate, w_up, ecnt, etok, epair, h_pair, 512, 0);
  // anchor experts: I=256 -> 4 n-tiles; 16 groups; h indexed by token
  gemm1_kernel<<<dim3(4, 64, NLG), 32, 0, stream>>>(
      x, aw_gate, aw_up, acnt, atok, nullptr, anchor_h, 256, 0);
  // shared expert: identity token mapping
  gemm1_kernel<<<dim3(8, 64, 1), 32, 0, stream>>>(
      x, sw_gate, sw_up, nullptr, nullptr, nullptr, shared_h, 512, TTOK);

  // down projections (N = 1024 -> 16 n-tiles of 64)
  gemm2_kernel<<<dim3(16, 64, NEXP), 32, 0, stream>>>(
      h_pair, w_down, ecnt, etok, epair, ew, routed_out, 512, 0, 1.0f);
  gemm2_kernel<<<dim3(16, 64, NLG), 32, 0, stream>>>(
      anchor_h, aw_down, acnt, atok, nullptr, nullptr, anchor_out, 256, 0, 0.1f);
  gemm2_kernel<<<dim3(16, 64, 1), 32, 0, stream>>>(
      shared_h, sw_down, nullptr, nullptr, nullptr, nullptr, shared_out, 512,
      TTOK, 1.0f);

  combine_kernel<<<(TTOK * HID) / 256, 256, 0, stream>>>(
      shared_out, anchor_out, routed_out, (float*)d_out);
}